// HippoSSKernel_12266426597443
// MI455X (gfx1250) — compile-verified
//
#include <hip/hip_runtime.h>
#include <hip/hip_bf16.h>
#include <math.h>

// ---------------------------------------------------------------------------
// HiPPO-LegS S4 SSKernel for MI455X (gfx1250, wave32).
//
// Stage 1 (hippo_kf_wmma): per (head h, freq f) compute the 2x2 complex block
//   r[a,b](h,f) = sum_n v[a,b,n]/(z_f - w_dt[n]) + conj(v[a,b,n])/(z_f - conj(w_dt[n]))
// as ONE real GEMM per head:  D(16x16f) = A(16x128) * B(128x16f)
//   - rows 0..7 of A: {re,im} of v[a,b] with dt folded in (rows 8..15 zero)
//   - K order = 4n+g, g in {recip_re, recip_im, recipc_re, recipc_im}
//   - B generated on the fly: z_f = 2i*tan(pi f/L)  (purely imaginary!)
// then Woodbury combine k_f = (r00 - r01*r10/(1+r11)) * (1 + i*tan(pi f/L)).
// Uses V_WMMA_F32_16X16X4_F32 (f32 matrix core; f16/bf16 would wreck the
// near-resonance reciprocals, so f32 WMMA is the right precision choice).
//
// Stage 2 (hippo_irfft): per-head irfft(L=16384) via the half-size
// complex-IFFT packing -> 8192-pt complex Stockham IFFT entirely in LDS.
// The 65.5KB half-spectrum is DMA'd into LDS by the Tensor Data Mover
// (tensor_load_to_lds, TENSORcnt), then the Hermitian pack runs LDS->LDS.
// Total LDS: 128KB ping-pong + 64KB spectrum = 192KB < CDNA5's 320KB/WGP.
// ---------------------------------------------------------------------------

#define H_   256
#define NH   32          // N/2 complex modes per head
#define L_   16384
#define F_   8193        // L/2 + 1
#define M_   8192        // half-size complex FFT length
#define KFB  (F_ * 2)    // floats per head in the k_f buffer (16386)
#define PI_F 3.14159265358979323846f

typedef __attribute__((ext_vector_type(2))) float v2f;
typedef __attribute__((ext_vector_type(8))) float v8f;
typedef __attribute__((ext_vector_type(4))) unsigned int u32x4;
typedef __attribute__((ext_vector_type(8))) int i32x8;
typedef __attribute__((ext_vector_type(4))) int i32x4;

// ---------------------------------------------------------------------------
// Stage 1: Cauchy kernel via f32 WMMA
// grid = (65, 256): 65 blocks of 8 waves -> 520 wave-tiles >= 513 tiles of 16
// freqs per head.
// ---------------------------------------------------------------------------
__global__ __launch_bounds__(256) void hippo_kf_wmma(
    const float* __restrict__ log_dt,
    const float* __restrict__ w_re, const float* __restrict__ w_im,
    const float* __restrict__ B_re, const float* __restrict__ B_im,
    const float* __restrict__ P_re, const float* __restrict__ P_im,
    const float* __restrict__ C_re, const float* __restrict__ C_im,
    float* __restrict__ kf)     // [H][F][2]
{
  // A matrix in LDS: layout [n(32)][m(16)][g(4)] so each lane fetches its
  // (VGPR0,VGPR1) A-fragment pair with one 8-byte conflict-free ds_load.
  __shared__ float Alds[NH * 16 * 4];        // 8 KB
  __shared__ float wdt_re[NH], wdt_im[NH];

  const int h   = blockIdx.y;
  const int tid = threadIdx.x;
  const float dt = __expf(log_dt[h]);

  if (tid < NH) {
    wdt_re[tid] = w_re[h * NH + tid] * dt;
    wdt_im[tid] = w_im[h * NH + tid] * dt;
  }
  if (tid < 4 * NH) {
    const int p = tid >> 5;       // (a,b) pair: p = 2a + b
    const int n = tid & 31;
    const int a = p >> 1, b = p & 1;
    // B_aug = [B, P],  C_aug = [C, conj(P)]
    float bar, bai, car, cai;
    if (a == 0) { bar = B_re[h * NH + n]; bai = B_im[h * NH + n]; }
    else        { bar = P_re[h * NH + n]; bai = P_im[h * NH + n]; }
    if (b == 0) { car = C_re[h * NH + n]; cai = C_im[h * NH + n]; }
    else        { car = P_re[h * NH + n]; cai = -P_im[h * NH + n]; }
    // v = B_aug * C_aug, with the trailing r*dt folded in
    const float vr = (bar * car - bai * cai) * dt;
    const float vi = (bar * cai + bai * car) * dt;
    float* base = &Alds[n * 64];
    const int mre = 2 * p, mim = 2 * p + 1;
    // re row: Re(v*recip) + Re(conj(v)*recip_c) = vr*rr - vi*ri + vr*rcr + vi*rci
    base[mre * 4 + 0] =  vr; base[mre * 4 + 1] = -vi;
    base[mre * 4 + 2] =  vr; base[mre * 4 + 3] =  vi;
    // im row: Im(v*recip) + Im(conj(v)*recip_c) = vi*rr + vr*ri - vi*rcr + vr*rci
    base[mim * 4 + 0] =  vi; base[mim * 4 + 1] =  vr;
    base[mim * 4 + 2] = -vi; base[mim * 4 + 3] =  vr;
    // rows 8..15 are padding -> zero
    float* zb = &Alds[n * 64 + (8 + 2 * p) * 4];
    #pragma unroll
    for (int g = 0; g < 8; ++g) zb[g] = 0.0f;
  }
  __syncthreads();

  const int wave   = tid >> 5;
  const int lane   = tid & 31;
  const int laneLo = lane & 15;
  const bool hi    = lane >= 16;

  const int tile = blockIdx.x * 8 + wave;   // wave-uniform
  const int f0   = tile * 16;
  if (f0 >= F_) return;                      // whole wave exits: EXEC all-1 inside

  const int   f   = f0 + laneLo;
  const float t   = __tanf(PI_F * (float)f * (1.0f / (float)L_));
  const float zim = 2.0f * t;                // z = 2i*tan(pi f / L)

  v8f acc = {};
  #pragma unroll 4
  for (int kk = 0; kk < NH; ++kk) {
    // A fragment: lanes 0-15 -> K=4kk+{0,1}, lanes 16-31 -> K=4kk+{2,3}
    const float2 af =
        *(const float2*)&Alds[(kk * 16 + laneLo) * 4 + (hi ? 2 : 0)];
    // B fragment: one complex reciprocal per lane.
    //   lanes 0-15 : recip   = 1/(z - w_dt)        -> rows rr, ri
    //   lanes 16-31: recip_c = 1/(z - conj(w_dt))  -> rows rcr, rci
    const float wr = wdt_re[kk], wi = wdt_im[kk];
    const float ar = -wr;
    const float bi = hi ? (zim + wi) : (zim - wi);
    const float inv = __builtin_amdgcn_rcpf(ar * ar + bi * bi);
    v2f Afrag = { af.x, af.y };
    v2f Bfrag = { ar * inv, -bi * inv };
    acc = __builtin_amdgcn_wmma_f32_16x16x4_f32(
        false, Afrag, false, Bfrag, (short)0, acc, false, false);
  }

  // Rows 0..7 (all we used) live in lanes 0..15 across acc[0..7]:
  // lane c holds r00re,r00im,r01re,r01im,r10re,r10im,r11re,r11im for freq f0+c.
  if (!hi && f < F_) {
    const float r00r = acc[0], r00i = acc[1];
    const float r01r = acc[2], r01i = acc[3];
    const float r10r = acc[4], r10i = acc[5];
    const float r11r = acc[6], r11i = acc[7];
    const float numr = r01r * r10r - r01i * r10i;
    const float numi = r01r * r10i + r01i * r10r;
    const float denr = 1.0f + r11r, deni = r11i;
    const float dinv = __builtin_amdgcn_rcpf(denr * denr + deni * deni);
    const float qr = (numr * denr + numi * deni) * dinv;
    const float qi = (numi * denr - numr * deni) * dinv;
    const float kr = r00r - qr, ki = r00i - qi;
    // k_f *= 2/(1+omega) = 1 + i*tan(pi f/L)
    float* dst = kf + ((size_t)h * F_ + f) * 2;
    dst[0] = kr - ki * t;
    dst[1] = ki + kr * t;
  }
}

// ---------------------------------------------------------------------------
// Stage 2: irfft(L) per head, one workgroup per head.
// Half-size packing: Y[k] = (S + D)/2, Y[M-k] = conj(S - D)/2 with
//   S = X[k] + conj(X[M-k]),  D = i*(X[k] - conj(X[M-k]))*e^{+i pi k/M},
// then y = IDFT_M(Y)/M, x[2m]=Re(y[m]), x[2m+1]=Im(y[m]).
// Stockham radix-2 (autosort, no bit reversal) in a 128KB LDS ping-pong.
// The half-spectrum is DMA'd into LDS by the TDM (one descriptor per block).
// ---------------------------------------------------------------------------
__global__ __launch_bounds__(256) void hippo_irfft(
    const float* __restrict__ kf,   // [H][F][2]
    float* __restrict__ out)        // [H][L]
{
  extern __shared__ float smem[];
  float2* bufA = (float2*)smem;            // M complex = 64 KB
  float2* bufB = bufA + M_;                // M complex = 64 KB
  float*  Xl   = (float*)(bufB + M_);      // KFB floats at byte offset 131072

  const int h   = blockIdx.x;
  const int tid = threadIdx.x;

  // ---- TDM: DMA the whole half-spectrum (65544 B) from global into LDS ----
  if (tid < 32) {   // wave 0 issues the tensor DMA (EXEC-independent per-wave op)
    const unsigned long long ga =
        (unsigned long long)(const void*)(kf + (size_t)h * KFB);
    // D# group 0: count=1 | lds_addr | global_addr[56:0] | type=2
    u32x4 g0;
    g0[0] = 1u;                                   // count=1 (valid descriptor)
    g0[1] = (unsigned)(2u * M_ * sizeof(float2)); // lds_addr = 131072 bytes
    g0[2] = (unsigned)(ga & 0xffffffffu);         // global_addr[31:0]
    g0[3] = (unsigned)((ga >> 32) & 0x01ffffffu)  // global_addr[56:32]
          | (2u << 30);                           // type = 2 ("image")
    // D# group 1: 1-D tile of 16386 4-byte elements
    i32x8 g1;
    g1[0] = (int)(2u << 16);            // workgroup_mask=0, data_size=2 (4B)
    g1[1] = (int)((unsigned)KFB << 16); // tensor_dim0[15:0] (bits 63:48)
    g1[2] = (int)(1u << 16);            // tensor_dim0[31:16]=0, tensor_dim1=1
    g1[3] = (int)((unsigned)KFB << 16); // tile_dim0 = 16386 (bits 127:112)
    g1[4] = 1;                          // tile_dim1 = 1, tile_dim2 = 0
    g1[5] = (int)KFB;                   // tensor_dim0_stride[31:0]
    g1[6] = 0;                          // stride0 hi, stride1 lo
    g1[7] = 0;
    i32x4 gz4 = { 0, 0, 0, 0 };               // groups 2/3 unused (<=2-D tensor)
    i32x8 gz8 = { 0, 0, 0, 0, 0, 0, 0, 0 };   // trailing group (clang-23 form)
    // 6-arg toolchain variant: (g0, g1, g2, g3, g4, cpol)
    __builtin_amdgcn_tensor_load_to_lds(g0, g1, gz4, gz4, gz8, 0);
    __builtin_amdgcn_s_wait_tensorcnt(0);
  }
  __syncthreads();

  // ---- pack half-spectrum into length-M complex sequence Y (LDS -> LDS) ----
  for (int k = tid; k <= M_ / 2; k += 256) {
    const float xr = Xl[2 * k],           xi = Xl[2 * k + 1];
    const float br = Xl[2 * (M_ - k)],    bi = -Xl[2 * (M_ - k) + 1]; // conj(X[M-k])
    const float sr = xr + br, si = xi + bi;
    const float dr = xr - br, di = xi - bi;
    float sw, cw;
    __sincosf(PI_F * (float)k * (1.0f / (float)M_), &sw, &cw); // e^{+i pi k/M}
    const float Dr = -di * cw - dr * sw;   // i*(dr+i di)*(cw+i sw)
    const float Di =  dr * cw - di * sw;
    bufA[k] = make_float2(0.5f * (sr + Dr), 0.5f * (si + Di));
    if (k > 0 && k < M_ / 2)
      bufA[M_ - k] = make_float2(0.5f * (sr - Dr), -0.5f * (si - Di));
  }
  __syncthreads();

  // ---- 13-stage inverse Stockham radix-2 ----
  float2* src = bufA;
  float2* dst = bufB;
  for (int s = 0; s < 13; ++s) {
    const int Ns = 1 << s;
    for (int t = tid; t < M_ / 2; t += 256) {
      const int q   = t & (Ns - 1);
      const int pNs = t - q;                 // p * Ns
      float sw, cw;
      __sincosf(PI_F * (float)q / (float)Ns, &sw, &cw);   // +sign: inverse
      const float2 a = src[t];
      const float2 b = src[t + M_ / 2];
      const float wbr = b.x * cw - b.y * sw;
      const float wbi = b.x * sw + b.y * cw;
      const int o = (pNs << 1) + q;
      dst[o]      = make_float2(a.x + wbr, a.y + wbi);
      dst[o + Ns] = make_float2(a.x - wbr, a.y - wbi);
    }
    __syncthreads();
    float2* tmp = src; src = dst; dst = tmp;
  }

  // ---- unpack: x[2m]=Re(y[m])/M, x[2m+1]=Im(y[m])/M ----
  const float scale = 1.0f / (float)M_;
  float* o = out + (size_t)h * L_;
  for (int m = tid; m < M_; m += 256) {
    const float2 y = src[m];
    o[2 * m]     = y.x * scale;
    o[2 * m + 1] = y.y * scale;
  }
}

// ---------------------------------------------------------------------------
extern "C" void kernel_launch(void* const* d_in, const int* in_sizes, int n_in,
                              void* d_out, int out_size, void* d_ws, size_t ws_size,
                              hipStream_t stream) {
  (void)in_sizes; (void)n_in; (void)out_size; (void)ws_size;
  const float* log_dt = (const float*)d_in[0];
  const float* w_re   = (const float*)d_in[1];
  const float* w_im   = (const float*)d_in[2];
  const float* B_re   = (const float*)d_in[3];
  const float* B_im   = (const float*)d_in[4];
  const float* P_re   = (const float*)d_in[5];
  const float* P_im   = (const float*)d_in[6];
  const float* C_re   = (const float*)d_in[7];
  const float* C_im   = (const float*)d_in[8];
  // d_in[9] is L (== 16384, compile-time LDEF)

  float* kf = (float*)d_ws;   // H * F * 2 floats = 16.78 MB scratch (fits L2)

  dim3 gridA(65, H_, 1);      // 65 blocks * 8 waves = 520 tiles >= 513
  hippo_kf_wmma<<<gridA, 256, 0, stream>>>(log_dt, w_re, w_im, B_re, B_im,
                                           P_re, P_im, C_re, C_im, kf);

  const size_t ldsB = 2 * M_ * sizeof(float2) + KFB * sizeof(float); // 196616 B
  hippo_irfft<<<H_, 256, ldsB, stream>>>(kf, (float*)d_out);
}